// GIN_14216341750214
// MI455X (gfx1250) — compile-verified
//
#include <hip/hip_runtime.h>
#include <hip/hip_bf16.h>
#include <math.h>

typedef __attribute__((ext_vector_type(2))) float v2f;
typedef __attribute__((ext_vector_type(8))) float v8f;

#define NN 50000
#define NE 800000
#define FIN 64
#define HD 128
#define NC 10
#define NG 512
#define BN_EPS 1e-5f

// ---------------------------------------------------------------- utility
__global__ void k_copy4(float4* __restrict__ o, const float4* __restrict__ in, long long n4) {
  long long i = (long long)blockIdx.x * blockDim.x + threadIdx.x;
  long long step = (long long)gridDim.x * blockDim.x;
  for (; i < n4; i += step) o[i] = in[i];
}

__global__ void k_zero(float* __restrict__ o, long long n) {
  long long i = (long long)blockIdx.x * blockDim.x + threadIdx.x;
  long long step = (long long)gridDim.x * blockDim.x;
  for (; i < n; i += step) o[i] = 0.0f;
}

// ------------------------------------------- edge scatter:  y[dst] += feat[src]
// one thread = one edge x one 16B feature chunk: 1 index pair, 1 b128 load, 4 atomics
template<int D>
__global__ void k_scatter_add(const float* __restrict__ feat, const int* __restrict__ srcv,
                              const int* __restrict__ dstv, float* __restrict__ y) {
  constexpr int C  = D / 4;                    // float4 chunks per edge
  constexpr int SH = (D == 64) ? 4 : 5;        // log2(C)
  long long total = (long long)NE * C;
  long long i = (long long)blockIdx.x * blockDim.x + threadIdx.x;
  long long step = (long long)gridDim.x * blockDim.x;
  for (; i < total; i += step) {
    int e = (int)(i >> SH);
    int c = (int)(i & (C - 1));
    int s = srcv[e], d = dstv[e];
    float4 v = ((const float4*)(feat + (long long)s * D))[c];
    float* p = y + (long long)d * D + c * 4;
    atomicAdd(p + 0, v.x);
    atomicAdd(p + 1, v.y);
    atomicAdd(p + 2, v.z);
    atomicAdd(p + 3, v.w);
  }
}

// ------------------------------- fused GIN MLP: relu(relu(bn(y@W1+b1))@W2+b2)
// block = 256 threads = 8 waves, handles 64 node rows (4 x 16-row subtiles).
// wave w owns output columns [16w,16w+16); each B fragment feeds 4 WMMAs.
template<int D_IN>
__global__ __launch_bounds__(256) void k_gin_mlp(
    const float* __restrict__ y,
    const float* __restrict__ w1, const float* __restrict__ b1,
    const float* __restrict__ gam, const float* __restrict__ bet,
    const float* __restrict__ mean, const float* __restrict__ var,
    const float* __restrict__ w2, const float* __restrict__ b2,
    float* __restrict__ out) {
  constexpr int ROWS = 64;
  __shared__ float sA[ROWS][D_IN + 1];   // +1: kill 16-way bank conflicts
  __shared__ float sT[ROWS][HD + 1];

  const int row0 = blockIdx.x * ROWS;
  const int lane = threadIdx.x & 31;
  const int wave = threadIdx.x >> 5;           // 0..7
  const int r    = lane & 15;                  // A-matrix row within subtile
  const int koff = (lane >> 4) * 2;            // lanes 0-15: K {0,1}; 16-31: K {2,3}
  const int ncol = wave * 16 + (lane & 15);    // output column 0..127
  const int mh   = (lane >> 4) * 8;            // C/D row offset

  // stage input tile (64 x D_IN) into LDS, float4 global loads
  for (int idx = threadIdx.x; idx < ROWS * (D_IN / 4); idx += 256) {
    int rr = idx / (D_IN / 4), c4 = idx % (D_IN / 4);
    int gr = row0 + rr;
    float4 v = make_float4(0.f, 0.f, 0.f, 0.f);
    if (gr < NN) v = ((const float4*)(y + (long long)gr * D_IN))[c4];
    sA[rr][c4 * 4 + 0] = v.x;
    sA[rr][c4 * 4 + 1] = v.y;
    sA[rr][c4 * 4 + 2] = v.z;
    sA[rr][c4 * 4 + 3] = v.w;
  }
  __syncthreads();

  // ---- GEMM 1: (64 x D_IN) @ (D_IN x 128), B fragment reused across 4 subtiles
  v8f acc0 = {}, acc1 = {}, acc2 = {}, acc3 = {};
  for (int k = 0; k < D_IN; k += 4) {
    v2f b, a0, a1, a2, a3;
    b.x  = w1[(k + koff) * HD + ncol];
    b.y  = w1[(k + koff + 1) * HD + ncol];
    a0.x = sA[r +  0][k + koff]; a0.y = sA[r +  0][k + koff + 1];
    a1.x = sA[r + 16][k + koff]; a1.y = sA[r + 16][k + koff + 1];
    a2.x = sA[r + 32][k + koff]; a2.y = sA[r + 32][k + koff + 1];
    a3.x = sA[r + 48][k + koff]; a3.y = sA[r + 48][k + koff + 1];
    acc0 = __builtin_amdgcn_wmma_f32_16x16x4_f32(false, a0, false, b, (short)0, acc0, false, false);
    acc1 = __builtin_amdgcn_wmma_f32_16x16x4_f32(false, a1, false, b, (short)0, acc1, false, false);
    acc2 = __builtin_amdgcn_wmma_f32_16x16x4_f32(false, a2, false, b, (short)0, acc2, false, false);
    acc3 = __builtin_amdgcn_wmma_f32_16x16x4_f32(false, a3, false, b, (short)0, acc3, false, false);
  }
  // bias + BN(eval) + ReLU -> LDS
  {
    float inv = rsqrtf(var[ncol] + BN_EPS);
    float g = gam[ncol], be = bet[ncol], mu = mean[ncol], bb = b1[ncol];
    v8f* accs[4] = { &acc0, &acc1, &acc2, &acc3 };
#pragma unroll
    for (int t = 0; t < 4; t++) {
#pragma unroll
      for (int q = 0; q < 8; q++) {
        float v = (*accs[t])[q] + bb;
        v = (v - mu) * inv * g + be;
        sT[t * 16 + q + mh][ncol] = fmaxf(v, 0.0f);
      }
    }
  }
  __syncthreads();

  // ---- GEMM 2: (64 x 128) @ (128 x 128)
  acc0 = (v8f){}; acc1 = (v8f){}; acc2 = (v8f){}; acc3 = (v8f){};
  for (int k = 0; k < HD; k += 4) {
    v2f b, a0, a1, a2, a3;
    b.x  = w2[(k + koff) * HD + ncol];
    b.y  = w2[(k + koff + 1) * HD + ncol];
    a0.x = sT[r +  0][k + koff]; a0.y = sT[r +  0][k + koff + 1];
    a1.x = sT[r + 16][k + koff]; a1.y = sT[r + 16][k + koff + 1];
    a2.x = sT[r + 32][k + koff]; a2.y = sT[r + 32][k + koff + 1];
    a3.x = sT[r + 48][k + koff]; a3.y = sT[r + 48][k + koff + 1];
    acc0 = __builtin_amdgcn_wmma_f32_16x16x4_f32(false, a0, false, b, (short)0, acc0, false, false);
    acc1 = __builtin_amdgcn_wmma_f32_16x16x4_f32(false, a1, false, b, (short)0, acc1, false, false);
    acc2 = __builtin_amdgcn_wmma_f32_16x16x4_f32(false, a2, false, b, (short)0, acc2, false, false);
    acc3 = __builtin_amdgcn_wmma_f32_16x16x4_f32(false, a3, false, b, (short)0, acc3, false, false);
  }
  {
    float bb = b2[ncol];
    v8f* accs[4] = { &acc0, &acc1, &acc2, &acc3 };
#pragma unroll
    for (int t = 0; t < 4; t++) {
#pragma unroll
      for (int q = 0; q < 8; q++) {
        int gr = row0 + t * 16 + q + mh;
        if (gr < NN) out[(long long)gr * HD + ncol] = fmaxf((*accs[t])[q] + bb, 0.0f);
      }
    }
  }
}

// ------------------------------- global add pool (one layer, column offset off)
__global__ void k_pool_scatter(const float* __restrict__ h, const int* __restrict__ batch,
                               float* __restrict__ pool, int off) {
  long long total = (long long)NN * (HD / 4);
  long long i = (long long)blockIdx.x * blockDim.x + threadIdx.x;
  long long step = (long long)gridDim.x * blockDim.x;
  for (; i < total; i += step) {
    int n = (int)(i >> 5);
    int c = (int)(i & 31);
    int bg = batch[n];
    float4 v = ((const float4*)(h + (long long)n * HD))[c];
    float* p = pool + (long long)bg * (3 * HD) + off + c * 4;
    atomicAdd(p + 0, v.x);
    atomicAdd(p + 1, v.y);
    atomicAdd(p + 2, v.z);
    atomicAdd(p + 3, v.w);
  }
}

// ------------------------------- lin1: relu(pool[512,384] @ W[384,384] + b)
// grid = (32, 3): blockIdx.x -> 16-row tile, blockIdx.y -> 128-column group
__global__ __launch_bounds__(256) void k_lin1(
    const float* __restrict__ A, const float* __restrict__ W,
    const float* __restrict__ bias, float* __restrict__ out) {
  __shared__ float sA[16][3 * HD + 1];
  const int K = 3 * HD;                        // 384
  const int row0 = blockIdx.x * 16;
  const int lane = threadIdx.x & 31;
  const int wave = threadIdx.x >> 5;
  const int r    = lane & 15;
  const int koff = (lane >> 4) * 2;
  const int ncol = blockIdx.y * 128 + wave * 16 + (lane & 15);
  const int mh   = (lane >> 4) * 8;

  for (int idx = threadIdx.x; idx < 16 * (K / 4); idx += 256) {
    int rr = idx / (K / 4), c4 = idx % (K / 4);
    float4 v = ((const float4*)(A + (long long)(row0 + rr) * K))[c4];
    sA[rr][c4 * 4 + 0] = v.x;
    sA[rr][c4 * 4 + 1] = v.y;
    sA[rr][c4 * 4 + 2] = v.z;
    sA[rr][c4 * 4 + 3] = v.w;
  }
  __syncthreads();

  v8f acc = {};
  for (int k = 0; k < K; k += 4) {
    v2f a, b;
    a.x = sA[r][k + koff];
    a.y = sA[r][k + koff + 1];
    b.x = W[(k + koff) * K + ncol];
    b.y = W[(k + koff + 1) * K + ncol];
    acc = __builtin_amdgcn_wmma_f32_16x16x4_f32(false, a, false, b,
                                                (short)0, acc, false, false);
  }
  float bb = bias[ncol];
#pragma unroll
  for (int q = 0; q < 8; q++) {
    out[(long long)(row0 + q + mh) * K + ncol] = fmaxf(acc[q] + bb, 0.0f);
  }
}

// ------------------------------- lin2 + log_softmax, one wave per graph
__global__ void k_lin2_logsoftmax(const float* __restrict__ h,   // [G, 384]
                                  const float* __restrict__ w,   // [384, 10]
                                  const float* __restrict__ b,
                                  float* __restrict__ out) {     // [G, 10]
  const int K = 3 * HD;
  int wave = threadIdx.x >> 5;
  int lane = threadIdx.x & 31;
  int g = blockIdx.x * (blockDim.x >> 5) + wave;
  if (g >= NG) return;

  float acc = 0.0f;
  if (lane < NC) {
    const float* hr = h + (long long)g * K;
    for (int k = 0; k < K; k++) acc += hr[k] * w[k * NC + lane];
    acc += b[lane];
  }
  float v = (lane < NC) ? acc : -INFINITY;
  float m = v;
#pragma unroll
  for (int off = 16; off >= 1; off >>= 1) m = fmaxf(m, __shfl_xor(m, off, 32));
  float e = (lane < NC) ? expf(v - m) : 0.0f;
  float s = e;
#pragma unroll
  for (int off = 16; off >= 1; off >>= 1) s += __shfl_xor(s, off, 32);
  if (lane < NC) out[(long long)g * NC + lane] = v - m - logf(s);
}

// ---------------------------------------------------------------- launcher
extern "C" void kernel_launch(void* const* d_in, const int* in_sizes, int n_in,
                              void* d_out, int out_size, void* d_ws, size_t ws_size,
                              hipStream_t stream) {
  const float* x     = (const float*)d_in[0];
  const int*   ei    = (const int*)d_in[1];
  const int*   batch = (const int*)d_in[2];
  const int*   srcv  = ei;           // edge_index[0]
  const int*   dstv  = ei + NE;      // edge_index[1]
  const float* P[24];
  for (int i = 0; i < 24; i++) P[i] = (const float*)d_in[3 + i];
  const float* lin1_w = (const float*)d_in[27];
  const float* lin1_b = (const float*)d_in[28];
  const float* lin2_w = (const float*)d_in[29];
  const float* lin2_b = (const float*)d_in[30];

  float* ws    = (float*)d_ws;
  float* y     = ws;                                   // [N,128] (layer1 uses first N*64)
  float* hA    = y  + (long long)NN * HD;              // [N,128]
  float* hB    = hA + (long long)NN * HD;              // [N,128]
  float* pool  = hB + (long long)NN * HD;              // [G,384]
  float* hpool = pool + (long long)NG * 3 * HD;        // [G,384]

  dim3 b256(256);
  const int mlpGrid = (NN + 63) / 64;                  // 782

  k_zero<<<(NG * 3 * HD + 255) / 256, b256, 0, stream>>>(pool, (long long)NG * 3 * HD);

  // ---- layer 1 (D_IN = 64)
  k_copy4<<<2048, b256, 0, stream>>>((float4*)y, (const float4*)x, (long long)NN * FIN / 4);
  k_scatter_add<FIN><<<4096, b256, 0, stream>>>(x, srcv, dstv, y);
  k_gin_mlp<FIN><<<mlpGrid, b256, 0, stream>>>(y, P[0], P[1], P[2], P[3], P[4], P[5], P[6], P[7], hA);
  k_pool_scatter<<<2048, b256, 0, stream>>>(hA, batch, pool, 0);

  // ---- layer 2
  k_copy4<<<2048, b256, 0, stream>>>((float4*)y, (const float4*)hA, (long long)NN * HD / 4);
  k_scatter_add<HD><<<4096, b256, 0, stream>>>(hA, srcv, dstv, y);
  k_gin_mlp<HD><<<mlpGrid, b256, 0, stream>>>(y, P[8], P[9], P[10], P[11], P[12], P[13], P[14], P[15], hB);
  k_pool_scatter<<<2048, b256, 0, stream>>>(hB, batch, pool, HD);

  // ---- layer 3
  k_copy4<<<2048, b256, 0, stream>>>((float4*)y, (const float4*)hB, (long long)NN * HD / 4);
  k_scatter_add<HD><<<4096, b256, 0, stream>>>(hB, srcv, dstv, y);
  k_gin_mlp<HD><<<mlpGrid, b256, 0, stream>>>(y, P[16], P[17], P[18], P[19], P[20], P[21], P[22], P[23], hA);
  k_pool_scatter<<<2048, b256, 0, stream>>>(hA, batch, pool, 2 * HD);

  // ---- classifier head
  k_lin1<<<dim3(NG / 16, 3), b256, 0, stream>>>(pool, lin1_w, lin1_b, hpool);
  k_lin2_logsoftmax<<<NG / 8, b256, 0, stream>>>(hpool, lin2_w, lin2_b, (float*)d_out);
}